// DeformConv3D_Block_19997367730265
// MI455X (gfx1250) — compile-verified
//
#include <hip/hip_runtime.h>
#include <hip/hip_bf16.h>

typedef float v2f __attribute__((ext_vector_type(2)));
typedef float v4f __attribute__((ext_vector_type(4)));
typedef float v8f __attribute__((ext_vector_type(8)));

#define CIN    16
#define COUT   16
#define DD     8
#define HH     96
#define WW     96
#define DP     10
#define HP     98
#define WP     98
#define NTAP   27
#define KTOT   (CIN * NTAP)      // 432
#define NOFF   (3 * NTAP)        // 81 offset channels
#define VOX    (DD * HH * WW)    // 73728
#define XP_ELEMS (DP * HP * WP * CIN)   // 1,536,640 floats
#define TILES_W (WW / 16)        // 6
#define WAVES_PER_WG 4
#define EPS 1e-5f

// ---------------------------------------------------------------------------
// Kernel 0: pack x [CIN][D][H][W] -> zero-padded channels-last xp [DP][HP][WP][CIN]
// ---------------------------------------------------------------------------
__global__ __launch_bounds__(256) void pack_x(const float* __restrict__ x,
                                              float* __restrict__ xp) {
    const int idx = blockIdx.x * 256 + threadIdx.x;
    if (idx >= XP_ELEMS) return;
    const int c  = idx & 15;
    const int sp = idx >> 4;
    const int wp = sp % WP;
    const int hp = (sp / WP) % HP;
    const int dp = sp / (WP * HP);
    float v = 0.f;
    if (dp >= 1 && dp <= DD && hp >= 1 && hp <= HH && wp >= 1 && wp <= WW)
        v = x[((c * DD + (dp - 1)) * HH + (hp - 1)) * WW + (wp - 1)];
    xp[idx] = v;
}

// ---------------------------------------------------------------------------
// Kernel 1: fused offset-conv (WMMA) + trilinear deform sampling + deform GEMM
// (WMMA).  One wave32 handles 16 consecutive output voxels along W.
//
// WMMA f32 16x16x4 lane layouts (CDNA5 ISA 7.12.2):
//   A (16x4):  lanes 0-15 hold M=lane,    v[0]=K+0, v[1]=K+1
//              lanes16-31 hold M=lane-16, v[0]=K+2, v[1]=K+3
//   B (4x16):  lanes 0-15 hold N=lane,    v[0]=K+0, v[1]=K+1
//              lanes16-31 hold N=lane-16, v[0]=K+2, v[1]=K+3
//   D (16x16): vgpr r: lanes 0-15 -> M=r, N=lane; lanes 16-31 -> M=r+8.
// ---------------------------------------------------------------------------
__global__ __launch_bounds__(128) void deform_fused(
    const float* __restrict__ xp,
    const float* __restrict__ w_off,   // [81][432]  (k = c*27 + tap)
    const float* __restrict__ b_off,   // [81]
    const float* __restrict__ w_def,   // [16][432]
    float* __restrict__ out)           // pre-BN output [16][VOX]
{
    __shared__ int   s_ktab[KTOT];                      // im2col address table
    __shared__ float s_off[WAVES_PER_WG][96 * 16];      // offsets: [ch][voxel]
    __shared__ float s_smp[WAVES_PER_WG][2 * 16 * 16];  // sampled chunk: [(tl*16+c)][voxel]

    // Build the k -> relative-address table once per workgroup:
    //   k = c*27 + tap, tap = kd*9 + kh*3 + kw
    //   off(k) = ((kd*HP + kh)*WP + kw)*CIN + c
    for (int k = threadIdx.x; k < KTOT; k += 128) {
        const int c  = k / NTAP;
        const int tp = k - c * NTAP;
        const int kd = tp / 9;
        const int r9 = tp - kd * 9;
        const int kh = r9 / 3;
        const int kw = r9 - kh * 3;
        s_ktab[k] = ((kd * HP + kh) * WP + kw) * CIN + c;
    }

    const int lane = threadIdx.x & 31;
    const int wv   = threadIdx.x >> 5;
    const int half = lane >> 4;
    const int n    = lane & 15;

    const int tile = blockIdx.x * WAVES_PER_WG + wv;     // 0..4607, all full
    const int wt = tile % TILES_W;
    const int h  = (tile / TILES_W) % HH;
    const int d  = tile / (TILES_W * HH);
    const int w0 = wt * 16;

    __syncthreads();   // s_ktab ready

    // ---- Stage 1: offset conv as implicit GEMM  D[96x16] += A[96x432]*B[432x16]
    v8f oacc[6];
#pragma unroll
    for (int t = 0; t < 6; t++)
#pragma unroll
        for (int r = 0; r < 8; r++) oacc[t][r] = 0.f;

    const int bbase = ((d * HP + h) * WP + (w0 + n)) * CIN;  // per-lane B base

#pragma unroll 2
    for (int kk = 0; kk < KTOT; kk += 4) {
        const int kb = kk + half * 2;
        v2f bfrag;
        bfrag[0] = xp[bbase + s_ktab[kb]];
        bfrag[1] = xp[bbase + s_ktab[kb + 1]];
        // tiles 0..4: rows 0..79, always valid -> unconditional b64 loads
#pragma unroll
        for (int t = 0; t < 5; t++) {
            const v2f afrag = *(const v2f*)(w_off + (t * 16 + n) * KTOT + kb);
            oacc[t] = __builtin_amdgcn_wmma_f32_16x16x4_f32(
                false, afrag, false, bfrag, (short)0, oacc[t], false, false);
        }
        // tile 5: only row 80 (lane n==0) is a real channel
        {
            v2f afrag; afrag[0] = 0.f; afrag[1] = 0.f;
            if (n == 0) afrag = *(const v2f*)(w_off + 80 * KTOT + kb);
            oacc[5] = __builtin_amdgcn_wmma_f32_16x16x4_f32(
                false, afrag, false, bfrag, (short)0, oacc[5], false, false);
        }
    }
    // stage offsets (+bias) into LDS
#pragma unroll
    for (int t = 0; t < 6; t++) {
#pragma unroll
        for (int r = 0; r < 8; r++) {
            const int ch = t * 16 + r + half * 8;
            float v = oacc[t][r];
            if (ch < NOFF) v += b_off[ch];
            s_off[wv][ch * 16 + n] = v;
        }
    }
    __syncthreads();

    // ---- Stage 2+3: per 2-tap chunk: trilinear sample -> LDS -> WMMA GEMM
    v8f dacc;
#pragma unroll
    for (int r = 0; r < 8; r++) dacc[r] = 0.f;

    for (int tb = 0; tb < NTAP; tb += 2) {
        const int ntap = (NTAP - tb) >= 2 ? 2 : 1;
        const int tl = half;                 // local tap index for this lane
        if (tl < ntap) {
            const int tap = tb + tl;
            const int kd = tap / 9, r9 = tap - kd * 9, kh = r9 / 3, kw = r9 - kh * 3;
            float pd = (float)(d + kd)          + s_off[wv][tap * 16 + n];
            float ph = (float)(h + kh)          + s_off[wv][(NTAP + tap) * 16 + n];
            float pw = (float)(w0 + n + kw)     + s_off[wv][(2 * NTAP + tap) * 16 + n];
            pd = fminf(fmaxf(pd, 0.f), (float)(DP - 1));
            ph = fminf(fmaxf(ph, 0.f), (float)(HP - 1));
            pw = fminf(fmaxf(pw, 0.f), (float)(WP - 1));
            const int d0 = (int)floorf(pd); const int d1 = (d0 + 1 < DP - 1) ? d0 + 1 : DP - 1;
            const int h0 = (int)floorf(ph); const int h1 = (h0 + 1 < HP - 1) ? h0 + 1 : HP - 1;
            const int wq = (int)floorf(pw); const int w1 = (wq + 1 < WP - 1) ? wq + 1 : WP - 1;
            const float fd = pd - (float)d0, fh = ph - (float)h0, fw = pw - (float)wq;
            const float gd = 1.f - fd, gh = 1.f - fh, gw = 1.f - fw;
            int   idx8[8]; float wt8[8];
            idx8[0] = ((d0 * HP + h0) * WP + wq) * CIN; wt8[0] = gd * gh * gw;
            idx8[1] = ((d0 * HP + h0) * WP + w1) * CIN; wt8[1] = gd * gh * fw;
            idx8[2] = ((d0 * HP + h1) * WP + wq) * CIN; wt8[2] = gd * fh * gw;
            idx8[3] = ((d0 * HP + h1) * WP + w1) * CIN; wt8[3] = gd * fh * fw;
            idx8[4] = ((d1 * HP + h0) * WP + wq) * CIN; wt8[4] = fd * gh * gw;
            idx8[5] = ((d1 * HP + h0) * WP + w1) * CIN; wt8[5] = fd * gh * fw;
            idx8[6] = ((d1 * HP + h1) * WP + wq) * CIN; wt8[6] = fd * fh * gw;
            idx8[7] = ((d1 * HP + h1) * WP + w1) * CIN; wt8[7] = fd * fh * fw;
            v4f s0 = {0,0,0,0}, s1 = {0,0,0,0}, s2 = {0,0,0,0}, s3 = {0,0,0,0};
#pragma unroll
            for (int i = 0; i < 8; i++) {
                const v4f* cp = (const v4f*)(xp + idx8[i]);   // 16 channels, 4x b128
                const float w = wt8[i];
                s0 += w * cp[0]; s1 += w * cp[1]; s2 += w * cp[2]; s3 += w * cp[3];
            }
#pragma unroll
            for (int e = 0; e < 4; e++) {
                s_smp[wv][(tl * 16 +  0 + e) * 16 + n] = s0[e];
                s_smp[wv][(tl * 16 +  4 + e) * 16 + n] = s1[e];
                s_smp[wv][(tl * 16 +  8 + e) * 16 + n] = s2[e];
                s_smp[wv][(tl * 16 + 12 + e) * 16 + n] = s3[e];
            }
        }
        __syncthreads();

        // deform GEMM over this chunk: rows l = tl*16 + c  <->  k_global = c*27 + (tb+tl)
        const int nks = ntap * 4;
        for (int ks = 0; ks < nks; ks++) {
            v2f afrag, bfrag;
#pragma unroll
            for (int j = 0; j < 2; j++) {
                const int l  = ks * 4 + half * 2 + j;
                const int tp = l >> 4;
                const int c  = l & 15;
                const int kg = c * NTAP + tb + tp;
                afrag[j] = w_def[n * KTOT + kg];     // A row M = lane&15 = cout
                bfrag[j] = s_smp[wv][l * 16 + n];    // B col N = voxel
            }
            dacc = __builtin_amdgcn_wmma_f32_16x16x4_f32(
                false, afrag, false, bfrag, (short)0, dacc, false, false);
        }
        __syncthreads();
    }

    // ---- epilogue: pre-BN output, coalesced along W
    const int voxBase = (d * HH + h) * WW + w0;
#pragma unroll
    for (int r = 0; r < 8; r++) {
        const int ch = r + half * 8;
        out[ch * VOX + voxBase + n] = dacc[r];
    }
}

// ---------------------------------------------------------------------------
// Kernel 2: deterministic per-channel sum / sumsq (one block per channel)
// ---------------------------------------------------------------------------
__global__ __launch_bounds__(256) void bn_reduce(const float* __restrict__ out,
                                                 float* __restrict__ bn_acc) {
    __shared__ float ss[256];
    __shared__ float sq[256];
    const int c = blockIdx.x;
    const float* p = out + c * VOX;
    float s = 0.f, s2 = 0.f;
    for (int i = threadIdx.x; i < VOX; i += 256) {
        const float v = p[i];
        s += v; s2 += v * v;
    }
    ss[threadIdx.x] = s; sq[threadIdx.x] = s2;
    __syncthreads();
    for (int st = 128; st > 0; st >>= 1) {
        if ((int)threadIdx.x < st) {
            ss[threadIdx.x] += ss[threadIdx.x + st];
            sq[threadIdx.x] += sq[threadIdx.x + st];
        }
        __syncthreads();
    }
    if (threadIdx.x == 0) { bn_acc[c] = ss[0]; bn_acc[16 + c] = sq[0]; }
}

// ---------------------------------------------------------------------------
// Kernel 3: BatchNorm (batch stats) + ReLU, in place on d_out
// ---------------------------------------------------------------------------
__global__ __launch_bounds__(256) void bn_apply(float* __restrict__ out,
                                                const float* __restrict__ bn_acc,
                                                const float* __restrict__ gamma,
                                                const float* __restrict__ beta) {
    const int idx = blockIdx.x * 256 + threadIdx.x;
    if (idx >= COUT * VOX) return;
    const int c = idx / VOX;
    const float inv_n = 1.f / (float)VOX;
    const float mean  = bn_acc[c] * inv_n;
    const float var   = bn_acc[16 + c] * inv_n - mean * mean;
    const float inv   = rsqrtf(var + EPS);
    const float y = gamma[c] * (out[idx] - mean) * inv + beta[c];
    out[idx] = fmaxf(y, 0.f);
}

// ---------------------------------------------------------------------------
extern "C" void kernel_launch(void* const* d_in, const int* in_sizes, int n_in,
                              void* d_out, int out_size, void* d_ws, size_t ws_size,
                              hipStream_t stream) {
    const float* x      = (const float*)d_in[0];   // [16][8][96][96]
    const float* w_off  = (const float*)d_in[1];   // [81][16][3][3][3]
    const float* b_off  = (const float*)d_in[2];   // [81]
    const float* w_def  = (const float*)d_in[3];   // [16][16][3][3][3]
    const float* gamma  = (const float*)d_in[4];   // [16]
    const float* beta   = (const float*)d_in[5];   // [16]
    float* out = (float*)d_out;                    // [16][8][96][96]

    float* xp     = (float*)d_ws;                  // XP_ELEMS floats
    float* bn_acc = xp + XP_ELEMS;                 // 32 floats

    pack_x<<<(XP_ELEMS + 255) / 256, 256, 0, stream>>>(x, xp);

    const int nWG = (VOX / 16) / WAVES_PER_WG;     // 4608 tiles / 4 = 1152
    deform_fused<<<nWG, WAVES_PER_WG * 32, 0, stream>>>(xp, w_off, b_off, w_def, out);

    bn_reduce<<<COUT, 256, 0, stream>>>(out, bn_acc);
    bn_apply<<<(COUT * VOX + 255) / 256, 256, 0, stream>>>(out, bn_acc, gamma, beta);
}